// GNNModel_59665685676469
// MI455X (gfx1250) — compile-verified
//
#include <hip/hip_runtime.h>

typedef __attribute__((ext_vector_type(2))) float v2f;
typedef __attribute__((ext_vector_type(8))) float v8f;

// ---------------------------------------------------------------------------
// Degree count: deg[dst[e]] += 1  (self-loop +1 folded into rsqrt kernel)
// ---------------------------------------------------------------------------
__global__ void deg_kernel(const int* __restrict__ dst, float* __restrict__ deg, int nE) {
    int e = blockIdx.x * blockDim.x + threadIdx.x;
    if (e < nE) atomicAdd(&deg[dst[e]], 1.0f);
}

__global__ void rsqrt_kernel(float* __restrict__ deg, int n) {
    int i = blockIdx.x * blockDim.x + threadIdx.x;
    if (i < n) deg[i] = rsqrtf(deg[i] + 1.0f);  // +1 = self loop
}

// ---------------------------------------------------------------------------
// H[n, DOUT] = X[n, DIN] @ W[DIN, DOUT] via V_WMMA_F32_16X16X4_F32.
// Handles FULL 16-row tiles only (remainder rows go to gemm_tail_kernel; for
// N=100000 there is no remainder). One wave computes a 16-row tile across all
// DOUT columns (DOUT/16 accumulators). W staged TRANSPOSED in LDS ([col][k])
// so each lane's B pair {W[k][col], W[k+1][col]} is one contiguous 8-byte
// ds_load_b64 into an even-aligned VGPR pair. 4 waves / block.
// ---------------------------------------------------------------------------
template <int DIN, int DOUT>
__global__ __launch_bounds__(128) void gemm_wmma_kernel(
    const float* __restrict__ X, const float* __restrict__ W,
    float* __restrict__ H, int full_tiles)
{
    constexpr int NT = DOUT / 16;
    __shared__ float Wlds[DIN * DOUT];     // transposed: Wlds[col*DIN + k] = W[k][col]
    for (int i = threadIdx.x; i < DIN * DOUT; i += blockDim.x) {
        int k = i / DOUT;
        int c = i - k * DOUT;
        Wlds[c * DIN + k] = W[i];          // coalesced global read
    }
    __syncthreads();

    const int wave  = threadIdx.x >> 5;
    const int lane  = threadIdx.x & 31;
    const int mtile = blockIdx.x * 4 + wave;
    if (mtile >= full_tiles) return;       // wave-uniform: EXEC stays all-1s

    const int l16   = lane & 15;
    const int khalf = (lane >> 4) * 2;     // lanes 0-15 -> K{0,1}, 16-31 -> K{2,3}
    const int row   = mtile * 16 + l16;

    v8f acc[NT];
    const v8f vzero = {0.f, 0.f, 0.f, 0.f, 0.f, 0.f, 0.f, 0.f};
#pragma unroll
    for (int t = 0; t < NT; ++t) acc[t] = vzero;

    const float* __restrict__ xr = X + (size_t)row * DIN;
#pragma unroll 4
    for (int k = 0; k < DIN; k += 4) {
        v2f a = *(const v2f*)(xr + k + khalf);   // A 16x4 fragment (fp32)
#pragma unroll
        for (int t = 0; t < NT; ++t) {
            // B 4x16 fragment: contiguous pair in transposed LDS image
            v2f b = *(const v2f*)(Wlds + (t * 16 + l16) * DIN + k + khalf);
            acc[t] = __builtin_amdgcn_wmma_f32_16x16x4_f32(
                false, a, false, b, (short)0, acc[t], false, false);
        }
    }

    // C/D layout: VGPR v -> M = v + 8*(lane>=16), N = lane&15.
    // Full tile: unguarded coalesced stores (cols 0..15 across half-wave).
    const int mbase = mtile * 16 + ((lane >> 4) ? 8 : 0);
#pragma unroll
    for (int t = 0; t < NT; ++t) {
        const int col = t * 16 + l16;
#pragma unroll
        for (int v = 0; v < 8; ++v) {
            H[(size_t)(mbase + v) * DOUT + col] = acc[t][v];
        }
    }
}

// Scalar tail for rows [row0, nrows) when nrows % 16 != 0 (not launched here).
template <int DIN, int DOUT>
__global__ void gemm_tail_kernel(const float* __restrict__ X, const float* __restrict__ W,
                                 float* __restrict__ H, int row0, int nrows)
{
    int r = row0 + blockIdx.x;
    int c = threadIdx.x;
    if (r >= nrows || c >= DOUT) return;
    float s = 0.f;
    for (int k = 0; k < DIN; ++k) s += X[(size_t)r * DIN + k] * W[k * DOUT + c];
    H[(size_t)r * DOUT + c] = s;
}

// ---------------------------------------------------------------------------
// Edge scatter: agg[dst] += dis[src]*dis[dst] * h[src]   (one wave per edge)
// ---------------------------------------------------------------------------
template <int F>
__global__ void agg_kernel(const float* __restrict__ h, const int* __restrict__ src,
                           const int* __restrict__ dst, const float* __restrict__ dis,
                           float* __restrict__ agg, int nE)
{
    int gw   = (int)((blockIdx.x * (size_t)blockDim.x + threadIdx.x) >> 5);
    int lane = threadIdx.x & 31;
    if (gw >= nE) return;
    int s = src[gw], d = dst[gw];
    float c = dis[s] * dis[d];
    const float* __restrict__ hs = h + (size_t)s * F;
    float* __restrict__ ad = agg + (size_t)d * F;
#pragma unroll
    for (int f = lane; f < F; f += 32)
        atomicAdd(&ad[f], c * hs[f]);
}

// ---------------------------------------------------------------------------
// z = relu(agg + dis^2 * h  (self loop) + b)   — written in place over agg
// ---------------------------------------------------------------------------
template <int F, int LOG2F>
__global__ void finalize_kernel(float* __restrict__ agg, const float* __restrict__ h,
                                const float* __restrict__ dis, const float* __restrict__ b,
                                int n)
{
    size_t idx = blockIdx.x * (size_t)blockDim.x + threadIdx.x;
    if (idx >= (size_t)n * F) return;
    int i = (int)(idx >> LOG2F);
    int f = (int)(idx & (F - 1));
    float di = dis[i];
    float z = agg[idx] + di * di * h[idx] + b[f];
    agg[idx] = z > 0.f ? z : 0.f;
}

// ---------------------------------------------------------------------------
// out[i] = dot(z2[i, 0:32], Wfc) + bfc     (one wave per node, shfl reduce)
// ---------------------------------------------------------------------------
__global__ void fc_kernel(const float* __restrict__ z, const float* __restrict__ Wfc,
                          const float* __restrict__ bfc, float* __restrict__ out, int n)
{
    int gw   = (int)((blockIdx.x * (size_t)blockDim.x + threadIdx.x) >> 5);
    int lane = threadIdx.x & 31;
    if (gw >= n) return;
    float v = z[(size_t)gw * 32 + lane] * Wfc[lane];
#pragma unroll
    for (int off = 16; off > 0; off >>= 1)
        v += __shfl_down(v, off, 32);
    if (lane == 0) out[gw] = v + bfc[0];
}

// ---------------------------------------------------------------------------
extern "C" void kernel_launch(void* const* d_in, const int* in_sizes, int n_in,
                              void* d_out, int out_size, void* d_ws, size_t ws_size,
                              hipStream_t stream)
{
    const float* x   = (const float*)d_in[0];
    const int*   ei  = (const int*)d_in[1];
    const float* W1  = (const float*)d_in[2];
    const float* b1  = (const float*)d_in[3];
    const float* W2  = (const float*)d_in[4];
    const float* b2  = (const float*)d_in[5];
    const float* Wfc = (const float*)d_in[6];
    const float* bfc = (const float*)d_in[7];
    float* out = (float*)d_out;

    const int N = in_sizes[0] / 128;
    const int E = in_sizes[1] / 2;
    const int* src = ei;
    const int* dst = ei + E;

    // Workspace layout: zeroed regions first so one memset covers them.
    float* ws   = (float*)d_ws;
    float* dis  = ws;                         // N       (degree -> rsqrt)
    float* agg1 = dis  + N;                   // N*64    (zeroed, becomes z1)
    float* agg2 = agg1 + (size_t)N * 64;      // N*32    (zeroed, becomes z2)
    float* h1   = agg2 + (size_t)N * 32;      // N*64
    float* h2   = h1   + (size_t)N * 64;      // N*32

    hipMemsetAsync(ws, 0, (size_t)N * 97 * sizeof(float), stream);

    // Degrees and symmetric norm factors
    deg_kernel<<<(E + 255) / 256, 256, 0, stream>>>(dst, dis, E);
    rsqrt_kernel<<<(N + 255) / 256, 256, 0, stream>>>(dis, N);

    const int full = N >> 4;                  // full 16-row tiles
    const int rem  = N - (full << 4);         // leftover rows (0 for N=100000)
    const int gblk = (full + 3) / 4;

    // Layer 1
    if (full)
        gemm_wmma_kernel<128, 64><<<gblk, 128, 0, stream>>>(x, W1, h1, full);
    if (rem)
        gemm_tail_kernel<128, 64><<<rem, 64, 0, stream>>>(x, W1, h1, full << 4, N);
    agg_kernel<64><<<(E + 7) / 8, 256, 0, stream>>>(h1, src, dst, dis, agg1, E);
    finalize_kernel<64, 6><<<((size_t)N * 64 + 255) / 256, 256, 0, stream>>>(agg1, h1, dis, b1, N);

    // Layer 2
    if (full)
        gemm_wmma_kernel<64, 32><<<gblk, 128, 0, stream>>>(agg1, W2, h2, full);
    if (rem)
        gemm_tail_kernel<64, 32><<<rem, 32, 0, stream>>>(agg1, W2, h2, full << 4, N);
    agg_kernel<32><<<(E + 7) / 8, 256, 0, stream>>>(h2, src, dst, dis, agg2, E);
    finalize_kernel<32, 5><<<((size_t)N * 32 + 255) / 256, 256, 0, stream>>>(agg2, h2, dis, b2, N);

    // FC head
    fc_kernel<<<(N + 7) / 8, 256, 0, stream>>>(agg2, Wfc, bfc, out, N);
}